// GNNEncoder_2559800508812
// MI455X (gfx1250) — compile-verified
//
#include <hip/hip_runtime.h>

// ---------- WMMA vector types (wave32, fp32 WMMA 16x16x4) ----------
typedef float v2f __attribute__((ext_vector_type(2)));
typedef float v8f __attribute__((ext_vector_type(8)));

#define N_NODES 50000
#define N_EDGES 800000
#define D_IN    64
#define HIDDEN  128
#define OUT_DIM 128

// -------------------------------------------------------------------
// Scatter-aggregate: agg[dst] += feat[src]  (+ degree count on layer 1)
// One thread handles one edge x one float4 chunk of features.
// -------------------------------------------------------------------
template <int K, bool COUNT>
__global__ __launch_bounds__(256)
void sage_scatter(const float* __restrict__ feat,
                  const int*   __restrict__ src,
                  const int*   __restrict__ dst,
                  float*       __restrict__ agg,
                  float*       __restrict__ cnt)
{
    constexpr int QUADS = K / 4;                       // float4 chunks per edge
    unsigned gid = blockIdx.x * blockDim.x + threadIdx.x;
    unsigned total = (unsigned)N_EDGES * QUADS;
    if (gid >= total) return;

    unsigned e = gid / QUADS;
    unsigned q = gid % QUADS;

    int s = src[e];
    int d = dst[e];

    const float4 v = *(const float4*)(feat + (size_t)s * K + q * 4);
    float* p = agg + (size_t)d * K + q * 4;
    atomicAdd(p + 0, v.x);
    atomicAdd(p + 1, v.y);
    atomicAdd(p + 2, v.z);
    atomicAdd(p + 3, v.w);

    if (COUNT && q == 0) atomicAdd(&cnt[d], 1.0f);
}

// -------------------------------------------------------------------
// Dense SAGE layer: out = relu?( (agg/deg) @ Wl + self @ Wr + bias )
// One wave32 computes a 16x16 output tile via V_WMMA_F32_16X16X4_F32.
//
// fp32 WMMA layouts (ISA 7.12.2):
//   A 16x4:  lanes 0-15 -> M=lane, (K=k0,k0+1); lanes 16-31 -> M=lane-16, (K=k0+2,k0+3)
//   B 4x16:  lanes 0-15 -> N=lane, (K=k0,k0+1); lanes 16-31 -> N=lane-16, (K=k0+2,k0+3)
//   C/D 8xf32: vgpr r -> row r (lanes 0-15, N=lane) / row r+8 (lanes 16-31)
// -------------------------------------------------------------------
template <int K, bool RELU>
__global__ __launch_bounds__(256)
void sage_gemm(const float* __restrict__ agg,
               const float* __restrict__ cnt,
               const float* __restrict__ selfin,
               const float* __restrict__ Wl,     // [K, OUT_DIM] row-major
               const float* __restrict__ Wr,     // [K, OUT_DIM] row-major
               const float* __restrict__ bias,   // [OUT_DIM]
               float*       __restrict__ out)    // [N_NODES, OUT_DIM]
{
    const unsigned wave = (blockIdx.x * blockDim.x + threadIdx.x) >> 5;
    const int lane = threadIdx.x & 31;

    const int node_tile = wave >> 3;          // OUT_DIM/16 == 8 col tiles
    const int col_tile  = wave & 7;
    if (node_tile >= N_NODES / 16) return;    // wave-uniform (exact grid anyway)

    const int nodebase = node_tile * 16;
    const int colbase  = col_tile * 16;

    const int half = lane >> 4;               // 0: K+0/1 , 1: K+2/3
    const int lidx = lane & 15;               // row (A) / col (B)
    const int koff = half * 2;

    // mean normalization: 1 / max(deg, 1)
    const float deg = cnt[nodebase + lidx];
    const float inv = 1.0f / fmaxf(deg, 1.0f);

    const float* aggrow  = agg    + (size_t)(nodebase + lidx) * K + koff;
    const float* selfrow = selfin + (size_t)(nodebase + lidx) * K + koff;

    v8f acc = {};
#pragma unroll
    for (int k0 = 0; k0 < K; k0 += 4) {
        v2f a1; a1.x = aggrow[k0]      * inv;  a1.y = aggrow[k0 + 1]  * inv;
        v2f a2; a2.x = selfrow[k0];            a2.y = selfrow[k0 + 1];

        const float* bl = Wl + (size_t)(k0 + koff) * OUT_DIM + colbase + lidx;
        const float* br = Wr + (size_t)(k0 + koff) * OUT_DIM + colbase + lidx;
        v2f b1; b1.x = bl[0]; b1.y = bl[OUT_DIM];
        v2f b2; b2.x = br[0]; b2.y = br[OUT_DIM];

        acc = __builtin_amdgcn_wmma_f32_16x16x4_f32(false, a1, false, b1,
                                                    (short)0, acc, false, false);
        acc = __builtin_amdgcn_wmma_f32_16x16x4_f32(false, a2, false, b2,
                                                    (short)0, acc, false, false);
    }

    const float bv = bias[colbase + lidx];
    float* outp = out + (size_t)nodebase * OUT_DIM + colbase + lidx;
#pragma unroll
    for (int r = 0; r < 8; ++r) {
        float v = acc[r] + bv;
        if (RELU) v = fmaxf(v, 0.0f);
        outp[(size_t)(r + half * 8) * OUT_DIM] = v;
    }
}

// -------------------------------------------------------------------
// Host launcher
// inputs: x[50000,64] f32 | edge_index[2,800000] int | W1_l[64,128] |
//         W1_r[64,128] | b1[128] | W2_l[128,128] | W2_r[128,128] | b2[128]
// output: [50000,128] f32
// -------------------------------------------------------------------
extern "C" void kernel_launch(void* const* d_in, const int* in_sizes, int n_in,
                              void* d_out, int out_size, void* d_ws, size_t ws_size,
                              hipStream_t stream)
{
    const float* x    = (const float*)d_in[0];
    const int*   edge = (const int*)  d_in[1];
    const float* W1l  = (const float*)d_in[2];
    const float* W1r  = (const float*)d_in[3];
    const float* b1   = (const float*)d_in[4];
    const float* W2l  = (const float*)d_in[5];
    const float* W2r  = (const float*)d_in[6];
    const float* b2   = (const float*)d_in[7];
    float* out = (float*)d_out;
    float* ws  = (float*)d_ws;

    const int* src = edge;              // edge_index[0, :]
    const int* dst = edge + N_EDGES;    // edge_index[1, :]

    // workspace layout (floats)
    float* cnt  = ws;                                   // 50000 (pad to 50048)
    float* agg1 = ws + 50048;                           // 50000*64
    float* h    = agg1 + (size_t)N_NODES * D_IN;        // 50000*128
    float* agg2 = h    + (size_t)N_NODES * HIDDEN;      // 50000*128

    // zero cnt + agg1 (contiguous) and agg2
    hipMemsetAsync(cnt,  0, (50048 + (size_t)N_NODES * D_IN) * sizeof(float), stream);
    hipMemsetAsync(agg2, 0, (size_t)N_NODES * HIDDEN * sizeof(float), stream);

    // ---- layer 1 ----
    {
        unsigned total = (unsigned)N_EDGES * (D_IN / 4);          // 12.8M threads
        sage_scatter<D_IN, true><<<(total + 255) / 256, 256, 0, stream>>>(
            x, src, dst, agg1, cnt);
    }
    {
        int blocks = (N_NODES / 16) * (OUT_DIM / 16) / 8;         // 3125 blocks, 8 waves each
        sage_gemm<D_IN, true><<<blocks, 256, 0, stream>>>(
            agg1, cnt, x, W1l, W1r, b1, h);
    }

    // ---- layer 2 ----
    {
        unsigned total = (unsigned)N_EDGES * (HIDDEN / 4);        // 25.6M threads
        sage_scatter<HIDDEN, false><<<(total + 255) / 256, 256, 0, stream>>>(
            h, src, dst, agg2, nullptr);
    }
    {
        int blocks = (N_NODES / 16) * (OUT_DIM / 16) / 8;
        sage_gemm<HIDDEN, false><<<blocks, 256, 0, stream>>>(
            agg2, cnt, h, W2l, W2r, b2, out);
    }
}